// GCNEncoder_32959579030035
// MI455X (gfx1250) — compile-verified
//
#include <hip/hip_runtime.h>

#define IN_C  256
#define HID_C 128
#define OUT_C 64

typedef float v2f __attribute__((ext_vector_type(2)));
typedef float v8f __attribute__((ext_vector_type(8)));
typedef unsigned int v4u __attribute__((ext_vector_type(4)));
typedef int v4i __attribute__((ext_vector_type(4)));
typedef int v8i __attribute__((ext_vector_type(8)));

__device__ __forceinline__ void atomAddF32(float* p, float v) {
  // relaxed, device scope -> native global_atomic_add_f32 (no CAS loop)
  __hip_atomic_fetch_add(p, v, __ATOMIC_RELAXED, __HIP_MEMORY_SCOPE_AGENT);
}

// ---------------- degree / norm ----------------

__global__ void k_init_deg(float* __restrict__ deg, int N) {
  int i = blockIdx.x * blockDim.x + threadIdx.x;
  if (i < N) deg[i] = 1.0f;   // self-loop contributes 1 to every node's degree
}

__global__ void k_count_deg(const int* __restrict__ dst, float* __restrict__ deg, int E) {
  int i = blockIdx.x * blockDim.x + threadIdx.x;
  if (i < E) atomAddF32(&deg[dst[i]], 1.0f);
}

__global__ void k_rsqrt_inplace(float* __restrict__ d, int N) {
  int i = blockIdx.x * blockDim.x + threadIdx.x;
  if (i < N) d[i] = rsqrtf(d[i]);   // deg >= 1 always
}

__global__ void k_edge_norm(const int* __restrict__ src, const int* __restrict__ dst,
                            const float* __restrict__ dinv, float* __restrict__ norm, int E) {
  int i = blockIdx.x * blockDim.x + threadIdx.x;
  if (i < E) norm[i] = dinv[src[i]] * dinv[dst[i]];
}

// ---------------- WMMA f32 GEMM: Out[M,Nw] = A[M,K] @ W[K,Nw] ----------------
// Block = 8 waves. blockIdx.x -> 8 m-tiles (one per wave); blockIdx.y -> one
// 64-column n-group (NT=4 16x16 tiles per wave). The K x 64 panel of W is
// staged in LDS once per block via the Tensor Data Mover (wave 0 issues a 2D
// tensor_load_to_lds, waits TENSORcnt, block barriers). In the MAC loop the B
// fragments come from LDS (ds_load, immediate offsets) and A fragments are
// prefetched 4 k-steps at a time (clause of global_load_b64), so the
// v_wmma_f32_16x16x4_f32 stream is not gated on global latency.
//
// A 16x4 f32 layout : lanes 0-15 hold (K=k0,k0+1), lanes 16-31 hold (K=k0+2,k0+3), M = lane&15
// B 4x16 f32 layout : mirrored (N = lane&15, same K split)
// C/D 16x16 f32     : VGPR j -> row j (lanes 0-15) / row j+8 (lanes 16-31), N = lane&15

template<int NT>
__global__ void k_gemm_wmma_f32(const float* __restrict__ A, const float* __restrict__ W,
                                float* __restrict__ Out, int M, int K, int Nw) {
  extern __shared__ float lds[];           // K x 64 panel, row-major, stride 64

  const int lane  = threadIdx.x & 31;
  const int wave  = threadIdx.x >> 5;
  const int nbase = blockIdx.y * 64;

  // ---- stage W[:, nbase:nbase+64] into LDS ----
#if __has_builtin(__builtin_amdgcn_tensor_load_to_lds)
  if (wave == 0) {
    const unsigned lds_base = (unsigned)__builtin_amdgcn_groupstaticsize(); // dynamic LDS starts here
    const unsigned long long ga = (unsigned long long)(const char*)(W + nbase);
    v4u g0;
    g0.x = 1u;                                   // count=1, user descriptor
    g0.y = lds_base;                             // LDS byte address
    g0.z = (unsigned)ga;                         // global addr [31:0]
    g0.w = (unsigned)((ga >> 32) & 0x01FFFFFFu) | (2u << 30);  // addr[56:32] | type=2
    v8i g1;
    g1[0] = 0x00020000;                          // wg_mask=0, data_size=2 (4B), no pad/iterate
    g1[1] = (Nw & 0xFFFF) << 16;                 // tensor_dim0 lo16 (barrier addr = 0)
    g1[2] = ((Nw >> 16) & 0xFFFF) | ((K & 0xFFFF) << 16);   // dim0 hi | tensor_dim1 lo
    g1[3] = ((K >> 16) & 0xFFFF) | (64 << 16);   // dim1 hi | tile_dim0 = 64
    g1[4] = (K & 0xFFFF);                        // tile_dim1 = K, tile_dim2 = 0 (2D)
    g1[5] = Nw;                                  // tensor_dim0_stride lo32
    g1[6] = 0;
    g1[7] = 0;
    v4i g2 = {0, 0, 0, 0};                       // unused (2D tensor)
    v4i g3 = {0, 0, 0, 0};
    v8i g4 = {0, 0, 0, 0, 0, 0, 0, 0};           // trailing group (clang-23 6-arg form)
    __builtin_amdgcn_tensor_load_to_lds(g0, g1, g2, g3, g4, 0);
    __builtin_amdgcn_s_wait_tensorcnt((short)0);
  }
#else
  for (int idx = threadIdx.x; idx < K * 64; idx += blockDim.x) {
    int k = idx >> 6, c = idx & 63;
    lds[k * 64 + c] = W[(size_t)k * Nw + nbase + c];
  }
#endif
  __syncthreads();

  const int mt = blockIdx.x * 8 + wave;
  if (mt * 16 >= M) return;          // uniform per wave: EXEC all-ones for WMMA

  const int half = lane >> 4;        // 0: K pair (k0,k0+1), 1: (k0+2,k0+3)
  const int l    = lane & 15;

  const float* arow = A + (size_t)(mt * 16 + l) * K + half * 2;
  const float* ldsb = lds + half * 2 * 64 + l;   // pre-offset B reads by lane's K-pair

  v8f acc[NT] = {};

  for (int k0 = 0; k0 < K; k0 += 16) {
    v2f a[4];
#pragma unroll
    for (int u = 0; u < 4; ++u)
      a[u] = *(const v2f*)(arow + k0 + 4 * u);   // clause of 4 global_load_b64
#pragma unroll
    for (int u = 0; u < 4; ++u) {
      const int ka = (k0 + 4 * u) * 64;
#pragma unroll
      for (int t = 0; t < NT; ++t) {
        v2f b;
        b.x = ldsb[ka + t * 16];                 // W[ka+2*half  ][nbase+t*16+l]
        b.y = ldsb[ka + 64 + t * 16];            // W[ka+2*half+1][nbase+t*16+l]
        acc[t] = __builtin_amdgcn_wmma_f32_16x16x4_f32(
            /*neg_a=*/false, a[u], /*neg_b=*/false, b,
            /*c_mod=*/(short)0, acc[t], /*reuse_a=*/false, /*reuse_b=*/false);
      }
    }
  }

  const int row0 = mt * 16 + half * 8;
#pragma unroll
  for (int t = 0; t < NT; ++t) {
    float* op = Out + (size_t)row0 * Nw + nbase + t * 16 + l;
#pragma unroll
    for (int j = 0; j < 8; ++j) op[(size_t)j * Nw] = acc[t][j];
  }
}

// ---------------- self-loop init / bias+relu ----------------

__global__ void k_self_init_hid(const float* __restrict__ h, const float* __restrict__ dinv,
                                float* __restrict__ agg, int total) {
  int i = blockIdx.x * blockDim.x + threadIdx.x;
  if (i < total) {
    int n = i >> 7;                  // / HID_C
    float dn = dinv[n];
    agg[i] = h[i] * dn * dn;         // self-loop edge: norm = dinv[n]^2
  }
}

__global__ void k_bias_relu_inplace(float* __restrict__ t, const float* __restrict__ b, int total) {
  int i = blockIdx.x * blockDim.x + threadIdx.x;
  if (i < total) {
    float v = t[i] + b[i & (HID_C - 1)];
    t[i] = v > 0.0f ? v : 0.0f;
  }
}

__global__ void k_self_init_out(const float* __restrict__ h, const float* __restrict__ dinv,
                                const float* __restrict__ b2, float* __restrict__ out, int total) {
  int i = blockIdx.x * blockDim.x + threadIdx.x;
  if (i < total) {
    int n = i >> 6;                  // / OUT_C
    float dn = dinv[n];
    out[i] = h[i] * dn * dn + b2[i & (OUT_C - 1)];
  }
}

// ---------------- edge scatter ----------------

// wave per edge, 128 channels: float4 per lane (b128 gather), 4 atomics/lane
__global__ void k_scatter128(const float* __restrict__ h, const int* __restrict__ src,
                             const int* __restrict__ dst, const float* __restrict__ norm,
                             float* __restrict__ agg, int E) {
  const int lane = threadIdx.x & 31;
  const int e = (blockIdx.x * blockDim.x + threadIdx.x) >> 5;
  if (e >= E) return;
  const int   s = src[e];
  const int   d = dst[e];
  const float w = norm[e];
  const float4 v = ((const float4*)(h + (size_t)s * HID_C))[lane];
  float* o = agg + (size_t)d * HID_C + lane * 4;
  atomAddF32(o + 0, v.x * w);
  atomAddF32(o + 1, v.y * w);
  atomAddF32(o + 2, v.z * w);
  atomAddF32(o + 3, v.w * w);
}

// half-wave (16 lanes) per edge, 64 channels: float4 per lane
__global__ void k_scatter64(const float* __restrict__ h, const int* __restrict__ src,
                            const int* __restrict__ dst, const float* __restrict__ norm,
                            float* __restrict__ out, int E) {
  const int t = blockIdx.x * blockDim.x + threadIdx.x;
  const int e = t >> 4;
  const int sl = t & 15;
  if (e >= E) return;
  const int   s = src[e];
  const int   d = dst[e];
  const float w = norm[e];
  const float4 v = ((const float4*)(h + (size_t)s * OUT_C))[sl];
  float* o = out + (size_t)d * OUT_C + sl * 4;
  atomAddF32(o + 0, v.x * w);
  atomAddF32(o + 1, v.y * w);
  atomAddF32(o + 2, v.z * w);
  atomAddF32(o + 3, v.w * w);
}

// ---------------- host-side orchestration ----------------

extern "C" void kernel_launch(void* const* d_in, const int* in_sizes, int n_in,
                              void* d_out, int out_size, void* d_ws, size_t ws_size,
                              hipStream_t stream) {
  (void)n_in; (void)out_size; (void)ws_size;

  const float* x   = (const float*)d_in[0];   // [N, 256]
  const int*   ei  = (const int*)  d_in[1];   // [2, E] row-major: [0]=src, [1]=dst
  const float* W1  = (const float*)d_in[2];   // [256, 128]
  const float* b1  = (const float*)d_in[3];   // [128]
  const float* W2  = (const float*)d_in[4];   // [128, 64]
  const float* b2  = (const float*)d_in[5];   // [64]
  float*       out = (float*)d_out;           // [N, 64]

  const int N = in_sizes[0] / IN_C;
  const int E = in_sizes[1] / 2;
  const int* src = ei;
  const int* dst = ei + E;

  // workspace carve-out (256B aligned slices)
  char* ws = (char*)d_ws;
  size_t off = 0;
  auto take = [&](size_t bytes) -> char* {
    char* p = ws + off;
    off = (off + bytes + 255) & ~(size_t)255;
    return p;
  };
  float* dinv = (float*)take((size_t)N * sizeof(float));
  float* norm = (float*)take((size_t)E * sizeof(float));
  float* h1   = (float*)take((size_t)N * HID_C * sizeof(float));
  float* agg  = (float*)take((size_t)N * HID_C * sizeof(float));
  float* h3   = (float*)take((size_t)N * OUT_C * sizeof(float));

  const int B = 256;
  auto cdiv = [](int a, int b) { return (a + b - 1) / b; };

  // ---- gcn_norm ----
  k_init_deg<<<cdiv(N, B), B, 0, stream>>>(dinv, N);
  k_count_deg<<<cdiv(E, B), B, 0, stream>>>(dst, dinv, E);
  k_rsqrt_inplace<<<cdiv(N, B), B, 0, stream>>>(dinv, N);
  k_edge_norm<<<cdiv(E, B), B, 0, stream>>>(src, dst, dinv, norm, E);

  const int mblocks = cdiv(N / 16, 8);        // 8 m-tiles (waves) per block

  // ---- layer 1: h1 = x @ W1 ----
  k_gemm_wmma_f32<4><<<dim3(mblocks, HID_C / 64), B, IN_C * 64 * sizeof(float), stream>>>(
      x, W1, h1, N, IN_C, HID_C);
  k_self_init_hid<<<cdiv(N * HID_C, B), B, 0, stream>>>(h1, dinv, agg, N * HID_C);
  k_scatter128<<<cdiv(E, B / 32), B, 0, stream>>>(h1, src, dst, norm, agg, E);
  k_bias_relu_inplace<<<cdiv(N * HID_C, B), B, 0, stream>>>(agg, b1, N * HID_C);

  // ---- layer 2: h3 = relu(agg+b1) @ W2 ----
  k_gemm_wmma_f32<4><<<dim3(mblocks, OUT_C / 64), B, HID_C * 64 * sizeof(float), stream>>>(
      agg, W2, h3, N, HID_C, OUT_C);
  k_self_init_out<<<cdiv(N * OUT_C, B), B, 0, stream>>>(h3, dinv, b2, out, N * OUT_C);
  k_scatter64<<<cdiv(E, B / 16), B, 0, stream>>>(h3, src, dst, norm, out, E);
}